// WavetableSynth_30039001268601
// MI455X (gfx1250) — compile-verified
//
#include <hip/hip_runtime.h>

// ---------------- problem constants (match reference) ----------------
constexpr int SRi   = 16000;
constexpr int Ltab  = 512;           // wavetable length (power of two)
constexpr int K     = 64;            // number of wavetables
constexpr int DUR   = 3;
constexpr int BLOCK = 160;           // attention upsample factor
constexpr int B     = 32;
constexpr int N     = SRi * DUR;     // 48000 samples per row
constexpr int J     = 100 * DUR;     // 300 attention columns
constexpr int JP    = 304;           // padded to 19 * 16 for WMMA tiling
constexpr float INC_SCALE = (float)Ltab / (float)SRi;

typedef __attribute__((ext_vector_type(2))) float v2f;
typedef __attribute__((ext_vector_type(8))) float v8f;

// ---------------- kernel 1: wts = [first4, tanh(rest)] ----------------
__global__ void k_prep_wts(const float* __restrict__ wavetables,
                           float* __restrict__ wts) {
  int i = blockIdx.x * blockDim.x + threadIdx.x;
  if (i >= K * Ltab) return;
  float x = wavetables[i];
  wts[i] = (i < 4 * Ltab) ? x : tanhf(x);
}

// ---------------- kernel 2: column softmax over K, zero-padded to JP --
__global__ void k_softmax(const float* __restrict__ attention,
                          float* __restrict__ att) {   // att: K x JP
  int j = blockIdx.x * blockDim.x + threadIdx.x;
  if (j >= JP) return;
  if (j >= J) {                       // pad columns -> zeros (M rows 300..303 = 0)
    for (int k = 0; k < K; ++k) att[k * JP + j] = 0.0f;
    return;
  }
  float m = -3.4e38f;
  for (int k = 0; k < K; ++k) m = fmaxf(m, attention[k * J + j]);
  float s = 0.0f;
  for (int k = 0; k < K; ++k) s += __expf(attention[k * J + j] - m);
  float inv = 1.0f / s;
  for (int k = 0; k < K; ++k) att[k * JP + j] = __expf(attention[k * J + j] - m) * inv;
}

// ---------------- kernel 3: M = att^T (JPx64) @ wts (64x512) ----------
// One wave32 per 16x16 output tile, K swept in steps of 4 via
// V_WMMA_F32_16X16X4_F32.  ISA layouts (7.12.2):
//   A 16x4  : lanes 0-15 M=0..15, v0=K0 v1=K1; lanes 16-31 v0=K2 v1=K3
//   B 4x16  : lanes 0-15 N=0..15, v0=K0 v1=K1; lanes 16-31 v0=K2 v1=K3
//   D 16x16 : VGPR r -> M=r (lanes 0-15) / M=r+8 (lanes 16-31), N=lane%16
constexpr int TILES_M = JP / 16;      // 19
constexpr int TILES_N = Ltab / 16;    // 32

__global__ void k_gemm(const float* __restrict__ att,   // K x JP
                       const float* __restrict__ wts,   // K x Ltab
                       float* __restrict__ M) {         // JP x Ltab
  const int gwave = (blockIdx.x * blockDim.x + threadIdx.x) >> 5;
  const int lane  = threadIdx.x & 31;
  const int tm = gwave / TILES_N;
  const int tn = gwave % TILES_N;
  if (tm >= TILES_M) return;          // wave-uniform; never taken with exact grid
  const int  l16 = lane & 15;
  const bool hi  = lane >= 16;
  const int arow = tm * 16 + l16;     // att column j == row of att^T
  const int bcol = tn * 16 + l16;     // wts column
  v8f acc = {};
  #pragma unroll
  for (int k0 = 0; k0 < K; k0 += 4) {
    const int ka = k0 + (hi ? 2 : 0);
    v2f a, b;
    a.x = att[(ka + 0) * JP + arow];
    a.y = att[(ka + 1) * JP + arow];
    b.x = wts[(ka + 0) * Ltab + bcol];
    b.y = wts[(ka + 1) * Ltab + bcol];
    // (neg_a, A, neg_b, B, c_mod, C, reuse_a, reuse_b)
    acc = __builtin_amdgcn_wmma_f32_16x16x4_f32(false, a, false, b,
                                                (short)0, acc, false, false);
  }
  const int mbase = tm * 16 + (hi ? 8 : 0);
  #pragma unroll
  for (int r = 0; r < 8; ++r)
    M[(size_t)(mbase + r) * Ltab + (tn * 16 + l16)] = acc[r];
}

// ---------------- decoupled scan: segment sums -> base scan -> synth --
constexpr int TPB  = 256;
constexpr int ELEM = 16;
constexpr int SEG  = TPB * ELEM;               // 4096 samples per segment
constexpr int NSEG = (N + SEG - 1) / SEG;      // 12 segments per row

__global__ void k_seg_sum(const float* __restrict__ pitch,
                          float* __restrict__ seg_sum) {
  __shared__ float red[TPB];
  const int b = blockIdx.x / NSEG;
  const int s = blockIdx.x % NSEG;
  const int tid = threadIdx.x;
  const float* prow = pitch + (size_t)b * N;
  const int n0 = s * SEG + tid * ELEM;
  float t = 0.0f;
  #pragma unroll
  for (int i = 0; i < ELEM; ++i) {
    int n = n0 + i;
    t += (n < N) ? prow[n] : 0.0f;
  }
  red[tid] = t * INC_SCALE;
  __syncthreads();
  for (int d = TPB >> 1; d > 0; d >>= 1) {
    if (tid < d) red[tid] += red[tid + d];
    __syncthreads();
  }
  if (tid == 0) seg_sum[blockIdx.x] = red[0];
}

__global__ void k_seg_scan(const float* __restrict__ seg_sum,
                           float* __restrict__ seg_base) {
  int b = threadIdx.x;
  if (b >= B) return;
  float base = 0.0f;
  for (int s = 0; s < NSEG; ++s) {
    float v = seg_sum[b * NSEG + s];
    seg_base[b * NSEG + s] = base;
    base += v;
  }
}

// Fused segment-local scan + synthesis:
//   out[b,n] = amp[b,n] * lerp(M[n/160, il], M[n/160, ih], alpha)
//   phase index = cumsum_b(inc) - inc[batch0]   (reference quirk), mod 512
__global__ void k_synth(const float* __restrict__ pitch,
                        const float* __restrict__ amplitude,
                        const float* __restrict__ M,
                        const float* __restrict__ seg_base,
                        float* __restrict__ out) {
  __shared__ float scan[TPB];
  const int b = blockIdx.x / NSEG;
  const int s = blockIdx.x % NSEG;
  const int tid = threadIdx.x;
  const float* prow = pitch + (size_t)b * N;
  const float* arow = amplitude + (size_t)b * N;
  float* orow = out + (size_t)b * N;
  const int n0 = s * SEG + tid * ELEM;

  // thread-local inclusive prefix of increments
  float inc[ELEM];
  float run = 0.0f;
  #pragma unroll
  for (int i = 0; i < ELEM; ++i) {
    int n = n0 + i;
    float f = (n < N) ? prow[n] : 0.0f;
    run += f * INC_SCALE;
    inc[i] = run;
  }

  // block-wide inclusive scan of per-thread totals (Hillis-Steele in LDS)
  scan[tid] = run;
  __syncthreads();
  for (int d = 1; d < TPB; d <<= 1) {
    float v = scan[tid];
    if (tid >= d) v += scan[tid - d];
    __syncthreads();
    scan[tid] = v;
    __syncthreads();
  }
  const float off   = (tid > 0) ? scan[tid - 1] : 0.0f;
  const float carry = seg_base[blockIdx.x] + off;

  #pragma unroll
  for (int i = 0; i < ELEM; ++i) {
    int n = n0 + i;
    if (n >= N) break;
    // subtract batch-0's increment row (reference: cumsum - increment[0])
    float idx = carry + inc[i] - pitch[n] * INC_SCALE;
    idx = fmodf(idx, (float)Ltab);
    if (idx < 0.0f) idx += (float)Ltab;
    int   il    = (int)idx;
    float alpha = idx - (float)il;
    il &= (Ltab - 1);
    int ih = (il + 1) & (Ltab - 1);
    const float* mrow = M + (size_t)(n / BLOCK) * Ltab;
    float lo = mrow[il];
    float hv = mrow[ih];
    orow[n] = (lo + alpha * (hv - lo)) * arow[n];
  }
}

// ---------------- launcher ----------------
extern "C" void kernel_launch(void* const* d_in, const int* in_sizes, int n_in,
                              void* d_out, int out_size, void* d_ws, size_t ws_size,
                              hipStream_t stream) {
  const float* pitch      = (const float*)d_in[0];   // B*N
  const float* amplitude  = (const float*)d_in[1];   // B*N
  const float* wavetables = (const float*)d_in[2];   // K*Ltab
  const float* attention  = (const float*)d_in[3];   // K*J

  float* ws       = (float*)d_ws;
  float* wts      = ws;                     // K*Ltab          (131072 B)
  float* att      = wts + K * Ltab;         // K*JP            ( 77824 B)
  float* Mmat     = att + K * JP;           // JP*Ltab         (622592 B)
  float* seg_sum  = Mmat + JP * Ltab;       // B*NSEG
  float* seg_base = seg_sum + B * NSEG;     // B*NSEG          total ~835 KB
  float* outp     = (float*)d_out;          // B*N floats

  k_prep_wts<<<(K * Ltab + 255) / 256, 256, 0, stream>>>(wavetables, wts);
  k_softmax<<<(JP + 63) / 64, 64, 0, stream>>>(attention, att);

  {
    const int waves  = TILES_M * TILES_N;           // 608 tiles, 1 wave each
    const int tpb    = 128;                          // 4 waves / block
    const int blocks = (waves * 32 + tpb - 1) / tpb; // 152 blocks exactly
    k_gemm<<<blocks, tpb, 0, stream>>>(att, wts, Mmat);
  }

  k_seg_sum<<<B * NSEG, TPB, 0, stream>>>(pitch, seg_sum);
  k_seg_scan<<<1, 32, 0, stream>>>(seg_sum, seg_base);
  k_synth<<<B * NSEG, TPB, 0, stream>>>(pitch, amplitude, Mmat, seg_base, outp);
}